// MaSIF_ppi_search_40613210751245
// MI455X (gfx1250) — compile-verified
//
#include <hip/hip_runtime.h>
#include <hip/hip_bf16.h>
#include <math.h>

#define S_DIM 256
#define V_DIM 200
#define NFEAT 5
#define NROT 16
#define G_DIM 80
#define NRHO 5
#define NTH 16
#define ROWS (NROT * S_DIM)          // 4096
#define TWO_PI_F 6.2831853071795864769f
#define EPS_F 1e-5f
#define VCH 8                        // v-chunk per barrier (200 % 8 == 0)

typedef __attribute__((ext_vector_type(2))) float v2f;
typedef __attribute__((ext_vector_type(8))) float v8f;

// ---------------------------------------------------------------------------
// Kernel B: fused gaussian weights + per-(r,s,g) normalization + einsum over v
// One block (96 threads) per (r,s).  Exploits the tiled mu/sigma structure:
// w[g] = tg[g%16] * rg[g/16] * mask  -> 21 exps per (r,s,v) instead of 160.
// v processed in chunks of 8: all 96 threads cooperatively stage
// 128 theta-exps + 40 rho-exps + 40 feats per chunk, then 80 lanes
// accumulate 8 v's per barrier pair (50 barriers total instead of 400).
// ---------------------------------------------------------------------------
__global__ __launch_bounds__(96) void gd_kernel(
    const float* __restrict__ feat,      // (S,V,5)
    const float* __restrict__ rho,       // (S,V)
    const float* __restrict__ theta,     // (S,V)
    const float* __restrict__ mask,      // (S,V,1)
    const float* __restrict__ mu_rho,    // (5,80) tiled
    const float* __restrict__ sig_rho,   // (5,80) tiled
    const float* __restrict__ mu_theta,  // (5,80) tiled
    const float* __restrict__ sig_theta, // (5,80) tiled
    float* __restrict__ GD)              // (5,4096,80)
{
    const int bi = blockIdx.x;           // 0..4095
    const int r  = bi >> 8;
    const int s  = bi & 255;
    const int t  = threadIdx.x;

    __shared__ float sTG[VCH][NTH];      // 128 theta gaussians
    __shared__ float sRG[VCH][NRHO];     // 40 rho gaussians (incl. mask)
    __shared__ float sF[VCH][NFEAT];     // 40 staged features
    __shared__ float sMu[NTH + NRHO];    // gaussian centers (16 theta + 5 rho)
    __shared__ float sIv[NTH + NRHO];    // 1/(sigma^2+eps)

    // Stage gaussian parameters once (row 0 of the f-tiled arrays; within a
    // rho-block of 16 g's the rho params are constant -> index i*16).
    if (t < NTH) {
        float sg = sig_theta[t];
        sMu[t] = mu_theta[t];
        sIv[t] = 1.f / (sg * sg + EPS_F);
    } else if (t < NTH + NRHO) {
        int i = t - NTH;
        float sg = sig_rho[i * NTH];
        sMu[t] = mu_rho[i * NTH];
        sIv[t] = 1.f / (sg * sg + EPS_F);
    }
    __syncthreads();

    const float rot  = (float)r * (TWO_PI_F / (float)NROT);
    const int   tid  = t & 15;           // theta index of g
    const int   rid  = t >> 4;           // rho index of g
    const int   base = s * V_DIM;

    float den = 0.f;
    float n0 = 0.f, n1 = 0.f, n2 = 0.f, n3 = 0.f, n4 = 0.f;

    for (int v0 = 0; v0 < V_DIM; v0 += VCH) {   // 25 chunks
        // Cooperative staging: flat idx 0..127 -> TG, 128..167 -> RG, 168..207 -> F
        for (int idx = t; idx < VCH * (NTH + NRHO + NFEAT); idx += 96) {
            if (idx < VCH * NTH) {
                int u = idx >> 4, j = idx & 15;
                float th = theta[base + v0 + u] + rot;
                if (th >= TWO_PI_F) th -= TWO_PI_F;  // mod 2pi (operands < 2pi)
                float d = th - sMu[j];
                sTG[u][j] = __expf(-d * d * sIv[j]);
            } else if (idx < VCH * (NTH + NRHO)) {
                int l = idx - VCH * NTH;
                int u = l / NRHO, i = l % NRHO;
                float d = rho[base + v0 + u] - sMu[NTH + i];
                sRG[u][i] = __expf(-d * d * sIv[NTH + i]) * mask[base + v0 + u];
            } else {
                int l = idx - VCH * (NTH + NRHO);
                int u = l / NFEAT, f = l % NFEAT;
                sF[u][f] = feat[(base + v0 + u) * NFEAT + f];
            }
        }
        __syncthreads();
        if (t < G_DIM) {
#pragma unroll
            for (int u = 0; u < VCH; ++u) {
                float w = sTG[u][tid] * sRG[u][rid];
                den += w;
                n0 += w * sF[u][0];
                n1 += w * sF[u][1];
                n2 += w * sF[u][2];
                n3 += w * sF[u][3];
                n4 += w * sF[u][4];
            }
        }
        __syncthreads();
    }

    if (t < G_DIM) {
        float inv = 1.f / (den + EPS_F);
        GD[(0 * ROWS + bi) * G_DIM + t] = n0 * inv;
        GD[(1 * ROWS + bi) * G_DIM + t] = n1 * inv;
        GD[(2 * ROWS + bi) * G_DIM + t] = n2 * inv;
        GD[(3 * ROWS + bi) * G_DIM + t] = n3 * inv;
        GD[(4 * ROWS + bi) * G_DIM + t] = n4 * inv;
    }
}

// ---------------------------------------------------------------------------
// Kernel C: cf[f] = gd[f] @ W_conv  via V_WMMA_F32_16X16X4_F32.
// One wave per 16x16 output tile; K=80 -> 20 wmma steps.
// A-frag layout (16x4 f32): lane = (k/2)*16 + m, vreg = k&1  => contiguous
// float2 per lane.  B-frag symmetric: lane = (k/2)*16 + n.
// ---------------------------------------------------------------------------
__global__ __launch_bounds__(32) void conv_gemm(
    const float* __restrict__ GD,        // (5,4096,80)
    const float* __restrict__ W,         // (80,80)
    float* __restrict__ CF)              // (5,4096,80)
{
    const int tile = blockIdx.x;         // 0..6399
    const int f   = tile / 1280;
    const int rem = tile % 1280;
    const int mt  = rem / 5;
    const int nt  = rem % 5;
    const int lane = threadIdx.x;
    const int m  = lane & 15;            // also column n for the B fragment
    const int kh = lane >> 4;

    const float* A = GD + (size_t)(f * ROWS + mt * 16) * G_DIM;

    v8f acc = {0.f, 0.f, 0.f, 0.f, 0.f, 0.f, 0.f, 0.f};
    for (int kk = 0; kk < G_DIM; kk += 4) {
        const int k0 = kk + 2 * kh;
        v2f a;
        a.x = A[m * G_DIM + k0];
        a.y = A[m * G_DIM + k0 + 1];
        v2f b;
        b.x = W[k0 * G_DIM + nt * 16 + m];
        b.y = W[(k0 + 1) * G_DIM + nt * 16 + m];
        acc = __builtin_amdgcn_wmma_f32_16x16x4_f32(
            false, a, false, b, (short)0, acc, false, false);
    }

    float* C = CF + (size_t)(f * ROWS + mt * 16) * G_DIM + nt * 16 + m;
#pragma unroll
    for (int i = 0; i < 8; ++i)
        C[(i + 8 * kh) * G_DIM] = acc[i];
}

// ---------------------------------------------------------------------------
// Kernel D: desc[s, f*80+g] = relu(max_r cf[f][r*256+s][g] + b_conv[f][g])
// ---------------------------------------------------------------------------
__global__ void maxrelu_kernel(
    const float* __restrict__ CF,        // (5,4096,80)
    const float* __restrict__ b_conv,    // (5,80)
    float* __restrict__ GPRE)            // (256,400)
{
    int idx = blockIdx.x * blockDim.x + threadIdx.x;
    if (idx >= S_DIM * NFEAT * G_DIM) return;
    const int s   = idx / (NFEAT * G_DIM);
    const int col = idx % (NFEAT * G_DIM);
    const int f = col / G_DIM;
    const int g = col % G_DIM;
    float mx = -INFINITY;
#pragma unroll
    for (int r = 0; r < NROT; ++r)
        mx = fmaxf(mx, CF[(size_t)(f * ROWS + r * S_DIM + s) * G_DIM + g]);
    mx += b_conv[f * G_DIM + g];
    GPRE[s * (NFEAT * G_DIM) + col] = fmaxf(mx, 0.f);
}

// ---------------------------------------------------------------------------
// Kernel E: global_desc = GPRE(256x400) @ fcc_w(400x80) + fcc_b  (WMMA f32)
// 16 M-tiles x 5 N-tiles = 80 waves, K=400 -> 100 wmma steps.
// ---------------------------------------------------------------------------
__global__ __launch_bounds__(32) void fcc_gemm(
    const float* __restrict__ A,         // (256,400)
    const float* __restrict__ Wf,        // (400,80)
    const float* __restrict__ bias,      // (80)
    float* __restrict__ out)             // (256,80) -> d_out[0..20479]
{
    const int tile = blockIdx.x;         // 0..79
    const int mt = tile / 5;
    const int nt = tile % 5;
    const int lane = threadIdx.x;
    const int m  = lane & 15;
    const int kh = lane >> 4;
    const int K = NFEAT * G_DIM;         // 400

    const float* Arow = A + (size_t)(mt * 16) * K;

    v8f acc = {0.f, 0.f, 0.f, 0.f, 0.f, 0.f, 0.f, 0.f};
    for (int kk = 0; kk < K; kk += 4) {
        const int k0 = kk + 2 * kh;
        v2f a;
        a.x = Arow[m * K + k0];
        a.y = Arow[m * K + k0 + 1];
        v2f b;
        b.x = Wf[k0 * G_DIM + nt * 16 + m];
        b.y = Wf[(k0 + 1) * G_DIM + nt * 16 + m];
        acc = __builtin_amdgcn_wmma_f32_16x16x4_f32(
            false, a, false, b, (short)0, acc, false, false);
    }

    const float bval = bias[nt * 16 + m];
#pragma unroll
    for (int i = 0; i < 8; ++i)
        out[(size_t)(mt * 16 + i + 8 * kh) * G_DIM + nt * 16 + m] = acc[i] + bval;
}

// ---------------------------------------------------------------------------
// Kernel F: triplet distances, score and data_loss.
// d_out layout: [0..20479] global_desc, [20480] data_loss, [20481..20608] score
// ---------------------------------------------------------------------------
__global__ __launch_bounds__(128) void loss_kernel(
    const float* __restrict__ gdesc,     // (256,80) (== d_out base)
    float* __restrict__ out)
{
    const int t = threadIdx.x;
    __shared__ float sh[128];

    const int i = t & 63;
    const float* a;
    const float* b;
    if (t < 64) { a = gdesc + (size_t)(64 + i) * G_DIM;  b = gdesc + (size_t)i * G_DIM; }          // binder - pos
    else        { a = gdesc + (size_t)(128 + i) * G_DIM; b = gdesc + (size_t)(192 + i) * G_DIM; }  // neg - neg2

    float acc = 0.f;
    for (int g = 0; g < G_DIM; ++g) {
        float d = a[g] - b[g];
        acc += d * d;
    }
    out[20481 + t] = acc;                // score: pos_d (0..63), neg_d (64..127)

    float rv = (t < 64) ? fmaxf(acc - 0.0f, 0.f) : fmaxf(10.0f - acc, 0.f);
    sh[t] = rv;
    __syncthreads();

    if (t == 0) {
        float mp = 0.f, mn = 0.f;
        for (int j = 0; j < 64; ++j) { mp += sh[j]; mn += sh[64 + j]; }
        mp *= (1.f / 64.f); mn *= (1.f / 64.f);
        float vp = 0.f, vn = 0.f;
        for (int j = 0; j < 64; ++j) {
            float dp = sh[j] - mp;      vp += dp * dp;
            float dn = sh[64 + j] - mn; vn += dn * dn;
        }
        vp *= (1.f / 63.f); vn *= (1.f / 63.f);   // ddof=1
        out[20480] = sqrtf(vp) + sqrtf(vn) + mp + mn;
    }
}

// ---------------------------------------------------------------------------
extern "C" void kernel_launch(void* const* d_in, const int* in_sizes, int n_in,
                              void* d_out, int out_size, void* d_ws, size_t ws_size,
                              hipStream_t stream) {
    const float* feat      = (const float*)d_in[0];
    const float* rho       = (const float*)d_in[1];
    const float* theta     = (const float*)d_in[2];
    const float* mask      = (const float*)d_in[3];
    const float* mu_rho    = (const float*)d_in[4];
    const float* sig_rho   = (const float*)d_in[5];
    const float* mu_theta  = (const float*)d_in[6];
    const float* sig_theta = (const float*)d_in[7];
    const float* W_conv    = (const float*)d_in[8];
    const float* b_conv    = (const float*)d_in[9];
    const float* fcc_w     = (const float*)d_in[10];
    const float* fcc_b     = (const float*)d_in[11];
    float* out = (float*)d_out;

    float* GD   = (float*)d_ws;                         // 5*4096*80 = 1,638,400 f
    float* CF   = GD + (size_t)NFEAT * ROWS * G_DIM;    // 1,638,400 f
    float* GPRE = CF + (size_t)NFEAT * ROWS * G_DIM;    // 256*400 = 102,400 f

    gd_kernel<<<ROWS, 96, 0, stream>>>(feat, rho, theta, mask,
                                       mu_rho, sig_rho, mu_theta, sig_theta, GD);
    conv_gemm<<<NFEAT * (ROWS / 16) * (G_DIM / 16), 32, 0, stream>>>(GD, W_conv, CF);
    maxrelu_kernel<<<(S_DIM * NFEAT * G_DIM + 255) / 256, 256, 0, stream>>>(CF, b_conv, GPRE);
    fcc_gemm<<<(S_DIM / 16) * (G_DIM / 16), 32, 0, stream>>>(GPRE, fcc_w, fcc_b, out);
    loss_kernel<<<1, 128, 0, stream>>>(out, out);
}